// IoU3DLossVariablePointHead_48644799595211
// MI455X (gfx1250) — compile-verified
//
#include <hip/hip_runtime.h>
#include <math.h>

#define TPB 256

#if defined(__gfx1250__) && __has_builtin(__builtin_amdgcn_global_load_async_to_lds_b32)
#define USE_ASYNC_LDS 1
#else
#define USE_ASYNC_LDS 0
#endif

#if USE_ASYNC_LDS
typedef __attribute__((address_space(1))) int as1_int;
typedef __attribute__((address_space(3))) int as3_int;
#endif

__device__ __forceinline__ float sdiv(float num, float den) {
    return (fabsf(den) > 1e-8f) ? (num / den) : 0.0f;
}

// Order-preserving float -> u32 map (IEEE total order for finite values)
__device__ __forceinline__ unsigned sortable_u32(float f) {
    unsigned u = __float_as_uint(f);
    return u ^ (((unsigned)((int)u >> 31)) | 0x80000000u);
}

// sortable_u32(500.0f) == 0xC3FA0000 ; valid points (angle < 500) compare below this
#define VALID_KEY_LIMIT 0xC3FA000000000000ull

__global__ __launch_bounds__(TPB) void IoU3DLossVariablePointHead_48644799595211_kernel(
    const float* __restrict__ gt, const float* __restrict__ pred,
    float* __restrict__ out, int n)
{
    __shared__ float s_gt[TPB * 7];
    __shared__ float s_pr[TPB * 7];

    const int tid    = threadIdx.x;
    const int base   = blockIdx.x * TPB;
    const int base7  = base * 7;
    const int total7 = n * 7;

    // Speculative prefetch of next block's tiles (global_prefetch_b8; OOB silently dropped)
    __builtin_prefetch(gt   + base7 + TPB * 7, 0, 1);
    __builtin_prefetch(pred + base7 + TPB * 7, 0, 1);

    // Coalesced AoS staging: 7 contiguous b32 sweeps per input array (ASYNCcnt path on gfx1250).
#pragma unroll
    for (int k = 0; k < 7; ++k) {
        int li = k * TPB + tid;
        int gi = base7 + li;
        gi = (gi < total7) ? gi : (total7 - 1);   // tail clamp (duplicate reads, harmless)
#if USE_ASYNC_LDS
        __builtin_amdgcn_global_load_async_to_lds_b32(
            (as1_int*)(gt + gi), (as3_int*)(&s_gt[li]), 0, 0);
        __builtin_amdgcn_global_load_async_to_lds_b32(
            (as1_int*)(pred + gi), (as3_int*)(&s_pr[li]), 0, 0);
#else
        s_gt[li] = gt[gi];
        s_pr[li] = pred[gi];
#endif
    }
#if USE_ASYNC_LDS
#if __has_builtin(__builtin_amdgcn_s_wait_asynccnt)
    __builtin_amdgcn_s_wait_asynccnt(0);
#else
    asm volatile("s_wait_asynccnt 0" ::: "memory");
#endif
#endif
    __syncthreads();

    const int i = base + tid;
    if (i >= n) return;

    const float* g = &s_gt[tid * 7];
    const float* p = &s_pr[tid * 7];
    float gw = g[0], glen = g[1], gh = g[2];
    float gx = g[3], gy = g[4], gz = g[5], gr = g[6];
    float pw = p[0], plen = p[1], ph = p[2];
    float qx = p[3], qy = p[4], qz = p[5], prr = p[6];

    float rxx = qx - gx, rxy = qy - gy;   // rel_xy
    float rr  = prr - gr;                 // rel_r

    float s1, c1;  __sincosf(rr,  &s1, &c1);
    float sg, cg;  __sincosf(gr,  &sg, &cg);
    float spp, cpp; __sincosf(prr, &spp, &cpp);

    // t1 = R(-gt_r) @ rel_xy ;  t2 = R(-pr_r) @ (-rel_xy)
    float t1x =  cg * rxx + sg * rxy;
    float t1y = -sg * rxx + cg * rxy;
    float t2x = -(cpp * rxx + spp * rxy);
    float t2y =  spp * rxx - cpp * rxy;

    float phw = 0.5f * pw, phl = 0.5f * plen;
    float ghw = 0.5f * gw, ghl = 0.5f * glen;

    const float vsx[4] = {  1.f, 1.f, -1.f, -1.f };
    const float vsy[4] = { -1.f, 1.f,  1.f, -1.f };

    float px[24], py[24], mk[24];

    // points[0..3] = rel_rot_pred_v = R(rel_r)@pred_verts + t1 ; mask = pred_in_gt
#pragma unroll
    for (int e = 0; e < 4; ++e) {
        float vx = vsx[e] * phw, vy = vsy[e] * phl;
        float ax = c1 * vx - s1 * vy + t1x;
        float ay = s1 * vx + c1 * vy + t1y;
        px[e] = ax; py[e] = ay;
        mk[e] = (fabsf(ax) <= ghw && fabsf(ay) <= ghl) ? 1.0f : 0.0f;
    }
    // points[4..7] = gt_v (unrotated) ; mask = gt_in_pred(rel_rot_gt_v = R(-rel_r)@gt_v + t2)
#pragma unroll
    for (int e = 0; e < 4; ++e) {
        float vx = vsx[e] * ghw, vy = vsy[e] * ghl;
        px[4 + e] = vx; py[4 + e] = vy;
        float bx_ =  c1 * vx + s1 * vy + t2x;
        float by_ = -s1 * vx + c1 * vy + t2y;
        mk[4 + e] = (fabsf(bx_) <= phw && fabsf(by_) <= phl) ? 1.0f : 0.0f;
    }
    // points[8..23] = edge/axis intersection candidates (4 per pred edge)
#pragma unroll
    for (int e = 0; e < 4; ++e) {
        int ep = (e + 3) & 3;                       // roll(v1, 1)
        float v1x = px[e],  v1y = py[e];
        float v0x = px[ep], v0y = py[ep];
        float dx = v1x - v0x, dy = v1y - v0y;
        float kx = sdiv(dy, dx);
        float bx = sdiv(v0y * v1x - v1y * v0x, dx);
        float ky = sdiv(dx, dy);
        float by = sdiv(v1y * v0x - v0y * v1x, dy);
        int o = 8 + e * 4;
        px[o + 0] =  ghw;            py[o + 0] =  kx * ghw + bx;
        px[o + 1] = -ghw;            py[o + 1] = -kx * ghw + bx;
        px[o + 2] =  ky * ghl + by;  py[o + 2] =  ghl;
        px[o + 3] = -ky * ghl + by;  py[o + 3] = -ghl;
    }
    // intersection-point masks: inside gt (slack 1e-3) AND inside pred (via R(-rel_r)@ip + t2)
#pragma unroll
    for (int t = 8; t < 24; ++t) {
        float ix = px[t], iy = py[t];
        float in_gt = (fabsf(ix) <= ghw + 1e-3f && fabsf(iy) <= ghl + 1e-3f) ? 1.0f : 0.0f;
        float rix =  c1 * ix + s1 * iy + t2x;
        float riy = -s1 * ix + c1 * iy + t2y;
        float in_pr = (fabsf(rix) <= phw + 1e-3f && fabsf(riy) <= phl + 1e-3f) ? 1.0f : 0.0f;
        mk[t] = in_gt * in_pr;
    }

    // masked centroid
    float msum = 0.f, sxm = 0.f, sym = 0.f;
#pragma unroll
    for (int t = 0; t < 24; ++t) { msum += mk[t]; sxm += px[t] * mk[t]; sym += py[t] * mk[t]; }
    float cx = sdiv(sxm, msum), cy = sdiv(sym, msum);
    float b0x = px[0] - cx, b0y = py[0] - cy;

    // Sort state: one sortable u64 key per point = (order-preserving angle bits << 32) | index.
    // Angle surrogate: diamond pseudo-angle of (dot+EPS, det+EPS) — strictly monotone in
    // atan2(det+EPS, dot+EPS), so the sorted order matches the reference's argsort; the
    // embedded index reproduces argsort's stability. Masked points get +1000 (sort after valid).
    unsigned long long key[32];
    float sxx[32], syy[32];
#pragma unroll
    for (int t = 0; t < 24; ++t) {
        float rx = px[t] - cx, ry = py[t] - cy;
        float dot = b0x * rx + b0y * ry + 1e-6f;   // x arg
        float det = b0x * ry - b0y * rx + 1e-6f;   // y arg
        float s = fabsf(dot) + fabsf(det);
        s = fmaxf(s, 1e-30f);
        float pa = copysignf(1.0f - dot / s, det);           // in [-2, 2]
        float keyf = pa + (1.0f - mk[t]) * 1000.0f;
        key[t] = ((unsigned long long)sortable_u32(keyf) << 32) | (unsigned)t;
        sxx[t] = px[t] * mk[t];           // sp * sm (masked points zeroed)
        syy[t] = py[t] * mk[t];
    }
#pragma unroll
    for (int t = 24; t < 32; ++t) { key[t] = 0xFFFFFFFFFFFFFFFFull; sxx[t] = 0.f; syy[t] = 0.f; }

    // Branchless bitonic network, 32 elems: one v_cmp_lt_u64 + cndmask swaps per CE.
    // Pad-pad comparators fold away (identical constant keys).
#pragma unroll
    for (int k = 2; k <= 32; k <<= 1) {
#pragma unroll
        for (int j = k >> 1; j > 0; j >>= 1) {
#pragma unroll
            for (int t = 0; t < 32; ++t) {
                int l = t ^ j;
                if (l > t) {
                    const bool up = ((t & k) == 0);
                    unsigned long long kt = key[t], kl = key[l];
                    bool sw = up ? (kt > kl) : (kt <= kl);
                    float xt = sxx[t], xl = sxx[l];
                    float yt = syy[t], yl = syy[l];
                    key[t] = sw ? kl : kt;  key[l] = sw ? kt : kl;
                    sxx[t] = sw ? xl : xt;  sxx[l] = sw ? xt : xl;
                    syy[t] = sw ? yl : yt;  syy[l] = sw ? yt : yl;
                }
            }
        }
    }

    // last valid sorted point (valid points form a sorted prefix; key < limit <=> angle < 500)
    float lvx = sxx[0], lvy = syy[0];
#pragma unroll
    for (int t = 0; t < 24; ++t) {
        bool val = key[t] < VALID_KEY_LIMIT;
        lvx = val ? sxx[t] : lvx;
        lvy = val ? syy[t] : lvy;
    }

    // shoelace over [last_v, s0..s23]
    float acc = lvx * syy[0] - sxx[0] * lvy;
#pragma unroll
    for (int t = 0; t < 23; ++t)
        acc += sxx[t] * syy[t + 1] - sxx[t + 1] * syy[t];
    float area = 0.5f * acc;

    // z overlap + IoU
    float top = fminf(gz + 0.5f * gh, qz + 0.5f * ph);
    float bot = fmaxf(gz - 0.5f * gh, qz - 0.5f * ph);
    float ih  = fmaxf(top - bot, 0.0f);
    float iv  = area * ih;
    float vg  = gw * glen * gh;
    float vp  = pw * plen * ph;
    out[i] = sdiv(iv, vg + vp - iv);
}

extern "C" void kernel_launch(void* const* d_in, const int* in_sizes, int n_in,
                              void* d_out, int out_size, void* d_ws, size_t ws_size,
                              hipStream_t stream) {
    const float* gt   = (const float*)d_in[0];
    const float* pred = (const float*)d_in[1];
    float* out = (float*)d_out;
    const int n = in_sizes[0] / 7;
    const int blocks = (n + TPB - 1) / TPB;
    hipLaunchKernelGGL(IoU3DLossVariablePointHead_48644799595211_kernel,
                       dim3(blocks), dim3(TPB), 0, stream, gt, pred, out, n);
}